// CapsuleLayer_78400333021447
// MI455X (gfx1250) — compile-verified
//
#include <hip/hip_runtime.h>
#include <hip/hip_bf16.h>

// Problem dims (fixed by reference): B=64, N=2048, I=16, J=32, O=32
#define Bsz 64
#define Nn  2048
#define Ii  16
#define Jj  32
#define Oo  32
#define JO  1024      // J*O columns of the per-n GEMM
#define NG  16        // n's handled per block

typedef float v2f __attribute__((ext_vector_type(2)));
typedef float v8f __attribute__((ext_vector_type(8)));

// ---------------------------------------------------------------------------
// Fused "routing pass" kernel (double-buffered async W staging).
//   pred[b, col] = sum_i x[b,n,i] * W[n, col, i]       (col = j*32+o)
//   FIRST:  c = 1/J (softmax of zeros)                 -> s += pred / J
//   !FIRST: logit[b,j] = sum_o pred[b,j,o]*vsum[b,j,o] -> c = softmax_j
//           s += c[b,j] * pred
// s accumulated in registers over NG n's, flushed with global f32 atomics.
// Block: 256 threads (8 waves). blockIdx.x = n-group, blockIdx.y = b-chunk(16).
// ---------------------------------------------------------------------------
template <bool FIRST>
__global__ __launch_bounds__(256, 1)
void routing_pass(const float* __restrict__ x,      // [B, N, I]
                  const float* __restrict__ w,      // [N, J, O, I]
                  const float* __restrict__ vsum,   // [B, J, O] (unused if FIRST)
                  float*       __restrict__ s_out)  // [B, J, O] accumulated
{
    __shared__ float lW[2][JO * Ii];     // 2 x 64 KB double buffer: W[n] as [col][i]
    __shared__ float lLogit[16 * Jj];    // 2 KB
    __shared__ float lC[16 * Jj];        // 2 KB

    const int tid  = threadIdx.x;
    const int lane = tid & 31;
    const int wv   = tid >> 5;           // wave id 0..7 -> owns 8 col-tiles
    const int bm   = lane & 15;          // row/col index inside a 16x16 tile
    const int hi   = lane >> 4;          // half-wave selector
    const int blb  = 8 * hi;             // local-b base for C/D VGPR rows
    const int bc   = blockIdx.y;         // b-chunk (16 rows)
    const int n0   = blockIdx.x * NG;

    // Issue one tile's async global->LDS copy: 16 b128 instructions per wave.
    auto issue_stage = [&](int n, int buf) {
        const float4* g4 = (const float4*)(w + (size_t)n * (JO * Ii));
        float4* l4 = (float4*)lW[buf];
#pragma unroll
        for (int i = 0; i < 16; ++i) {
            const int idx = tid + i * 256;
            const unsigned laddr = (unsigned)(uintptr_t)(l4 + idx);
            const float4* gaddr  = g4 + idx;
            asm volatile("global_load_async_to_lds_b128 %0, %1, off"
                         :: "v"(laddr), "v"(gaddr) : "memory");
        }
    };

    // s accumulators: 8 tiles x 8 rows
    v8f sacc[8];
#pragma unroll
    for (int t = 0; t < 8; ++t) sacc[t] = (v8f){0.f,0.f,0.f,0.f,0.f,0.f,0.f,0.f};

    // Per-lane vsum cache matching the C/D tile layout (b varies over VGPR rows)
    float vreg[8][8];
    if constexpr (!FIRST) {
#pragma unroll
        for (int tt = 0; tt < 8; ++tt) {
            const int col = (wv * 8 + tt) * 16 + bm;
#pragma unroll
            for (int r = 0; r < 8; ++r) {
                const int b = bc * 16 + blb + r;
                vreg[tt][r] = vsum[b * JO + col];
            }
        }
    }

    issue_stage(n0, 0);                                   // prime the pipeline

    for (int nn = 0; nn < NG; ++nn) {
        const int n   = n0 + nn;
        const int buf = nn & 1;

        // Overlap: kick off next tile's copy, then wait only for current tile
        // (async loads complete in order; leaving 16 outstanding == next tile).
        if (nn + 1 < NG) {
            issue_stage(n + 1, buf ^ 1);
            if (nn + 2 < NG)   // warm L2 for depth-2 tile
                __builtin_prefetch(w + (size_t)(n + 2) * (JO * Ii) + (size_t)tid * 64, 0, 1);
            asm volatile("s_wait_asynccnt 16" ::: "memory");
        } else {
            asm volatile("s_wait_asynccnt 0" ::: "memory");
        }
        __syncthreads();

        // ---- A fragments: 16x4 fp32, lanes 0-15 hold K 2h,2h+1 (h=hi) ----
        const float* xp = x + ((size_t)(bc * 16 + bm) * Nn + n) * Ii + 2 * hi;
        v2f A[4];
#pragma unroll
        for (int s = 0; s < 4; ++s) { A[s].x = xp[4 * s]; A[s].y = xp[4 * s + 1]; }

        // ---- WMMA: pred tile accumulation (K = 16 in 4 steps of 4) ----
        v8f pacc[8];
#pragma unroll
        for (int tt = 0; tt < 8; ++tt) {
            const int col = (wv * 8 + tt) * 16 + bm;     // this lane's N index
            const float* wp = &lW[buf][col * Ii + 2 * hi];
            v8f acc = (v8f){0.f,0.f,0.f,0.f,0.f,0.f,0.f,0.f};
#pragma unroll
            for (int s = 0; s < 4; ++s) {
                v2f Bf; Bf.x = wp[4 * s]; Bf.y = wp[4 * s + 1];
                acc = __builtin_amdgcn_wmma_f32_16x16x4_f32(
                        false, A[s], false, Bf, (short)0, acc, false, false);
            }
            pacc[tt] = acc;
        }

        if constexpr (FIRST) {
            // uniform coupling c = 1/J
#pragma unroll
            for (int tt = 0; tt < 8; ++tt)
#pragma unroll
                for (int r = 0; r < 8; ++r)
                    sacc[tt][r] += pacc[tt][r] * (1.0f / (float)Jj);
            __syncthreads();   // all lW[buf] reads done before it is refilled
        } else {
            // ---- agreement logits via LDS f32 atomics ----
            for (int i = tid; i < 16 * Jj; i += 256) lLogit[i] = 0.f;
            __syncthreads();
#pragma unroll
            for (int tt = 0; tt < 8; ++tt) {
                const int j = (wv * 8 + tt) >> 1;        // tile -> capsule j
#pragma unroll
                for (int r = 0; r < 8; ++r)
                    atomicAdd(&lLogit[(blb + r) * Jj + j], pacc[tt][r] * vreg[tt][r]);
            }
            __syncthreads();
            // ---- softmax over j, one thread per b-row ----
            if (tid < 16) {
                float mx = -3.4e38f;
                for (int j = 0; j < Jj; ++j) mx = fmaxf(mx, lLogit[tid * Jj + j]);
                float sum = 0.f;
                for (int j = 0; j < Jj; ++j) {
                    float e = __expf(lLogit[tid * Jj + j] - mx);
                    lC[tid * Jj + j] = e; sum += e;
                }
                const float inv = 1.0f / sum;
                for (int j = 0; j < Jj; ++j) lC[tid * Jj + j] *= inv;
            }
            __syncthreads();
            // ---- s += c * pred ----
#pragma unroll
            for (int tt = 0; tt < 8; ++tt) {
                const int j = (wv * 8 + tt) >> 1;
#pragma unroll
                for (int r = 0; r < 8; ++r)
                    sacc[tt][r] += lC[(blb + r) * Jj + j] * pacc[tt][r];
            }
            __syncthreads();   // all lW[buf]/lC reads done before refill
        }
    }

    // ---- flush register s-partials with global f32 atomics ----
#pragma unroll
    for (int tt = 0; tt < 8; ++tt) {
        const int col = (wv * 8 + tt) * 16 + bm;
#pragma unroll
        for (int r = 0; r < 8; ++r) {
            const int b = bc * 16 + blb + r;
            atomicAdd(&s_out[b * JO + col], sacc[tt][r]);
        }
    }
}

// ---------------------------------------------------------------------------
// squash: v = s * (|s|^2/(1+|s|^2)) / sqrt(|s|^2 + 1e-8), reduce over O=32.
// One wave per (b,j) row; accumulate into vsum or write final output.
// ---------------------------------------------------------------------------
__global__ __launch_bounds__(256)
void squash_kernel(const float* __restrict__ s, float* __restrict__ dst,
                   int accumulate)
{
    const int row = blockIdx.x * 8 + (threadIdx.x >> 5);  // 0 .. B*J-1
    const int o   = threadIdx.x & 31;
    const float val = s[row * Oo + o];
    float sq = val * val;
#pragma unroll
    for (int off = 16; off > 0; off >>= 1) sq += __shfl_xor(sq, off, 32);
    const float scale = sq / (1.0f + sq) * rsqrtf(sq + 1e-8f);
    const float v = val * scale;
    if (accumulate) dst[row * Oo + o] += v;
    else            dst[row * Oo + o]  = v;
}

__global__ void zero_kernel(float* __restrict__ p, int n)
{
    const int i = blockIdx.x * blockDim.x + threadIdx.x;
    if (i < n) p[i] = 0.f;
}

// ---------------------------------------------------------------------------
extern "C" void kernel_launch(void* const* d_in, const int* in_sizes, int n_in,
                              void* d_out, int out_size, void* d_ws, size_t ws_size,
                              hipStream_t stream)
{
    const float* x = (const float*)d_in[0];   // [64, 2048, 16]
    const float* w = (const float*)d_in[1];   // [2048, 32, 32, 16]
    float* out   = (float*)d_out;             // [64, 32, 32]
    float* s_buf = (float*)d_ws;              // 65536 floats
    float* vsum  = s_buf + (Bsz * JO);        // 65536 floats

    const dim3 grid(Nn / NG, Bsz / 16);       // (128, 4)
    const int SQ_BLOCKS = (Bsz * Jj) / 8;     // 256

    // zero s_buf and vsum
    zero_kernel<<<(2 * Bsz * JO + 255) / 256, 256, 0, stream>>>(s_buf, 2 * Bsz * JO);

    // iter 0: c uniform; v0 -> vsum
    routing_pass<true ><<<grid, 256, 0, stream>>>(x, w, nullptr, s_buf);
    squash_kernel<<<SQ_BLOCKS, 256, 0, stream>>>(s_buf, vsum, 1);

    // iter 1: logits = pred . v0 ; vsum += v1
    zero_kernel<<<(Bsz * JO + 255) / 256, 256, 0, stream>>>(s_buf, Bsz * JO);
    routing_pass<false><<<grid, 256, 0, stream>>>(x, w, vsum, s_buf);
    squash_kernel<<<SQ_BLOCKS, 256, 0, stream>>>(s_buf, vsum, 1);

    // iter 2: logits = pred . (v0+v1) ; final v2 -> out
    zero_kernel<<<(Bsz * JO + 255) / 256, 256, 0, stream>>>(s_buf, Bsz * JO);
    routing_pass<false><<<grid, 256, 0, stream>>>(x, w, vsum, s_buf);
    squash_kernel<<<SQ_BLOCKS, 256, 0, stream>>>(s_buf, out, 0);
}